// FilterDetection_61950608277598
// MI455X (gfx1250) — compile-verified
//
#include <hip/hip_runtime.h>

#define BB 4
#define NN 20000
#define CC 21
#define PP 100
#define NCLS (CC - 1)
#define SCORE_THR 0.7f
#define BIGF 1.0e9f
#define MAX_RATIO_F 4.135166556742356f
#define NT 256

// ---------------------------------------------------------------------------
// Kernel 1: le = score * logits, validity, box decode, per-class score rows
// ---------------------------------------------------------------------------
__global__ __launch_bounds__(NT) void fd_prep(
    const float* __restrict__ score,    // (B,N,1)
    const float* __restrict__ logits,   // (B,N,C)
    const float* __restrict__ regress,  // (B,N,4)
    const float* __restrict__ anchors,  // (N,4)
    float* __restrict__ boxArr,         // (B,N,4)
    float* __restrict__ clsArr)         // (B,NCLS,N)
{
    int t = blockIdx.x * blockDim.x + threadIdx.x;
    if (t >= BB * NN) return;
    int b = t / NN;
    int n = t - b * NN;

    float s = score[t];
    const float* lg = logits + (size_t)t * CC;
    float lev[CC];
    float best = -INFINITY;
    for (int c = 0; c < CC; ++c) {
        float v = s * lg[c];
        lev[c] = v;
        if (v > best) best = v;
    }
    // label>0  <=>  first argmax index != 0  <=>  lev[0] < best
    bool valid = (best >= SCORE_THR) && (lev[0] < best);

    // decode boxes
    const float* anc = anchors + (size_t)n * 4;
    float ax1 = anc[0], ay1 = anc[1], ax2 = anc[2], ay2 = anc[3];
    float aw = ax2 - ax1, ah = ay2 - ay1;
    float acx = ax1 + 0.5f * aw, acy = ay1 + 0.5f * ah;
    const float* rg = regress + (size_t)t * 4;
    float dx = rg[0], dy = rg[1];
    float dw = fminf(fmaxf(rg[2], -MAX_RATIO_F), MAX_RATIO_F);
    float dh = fminf(fmaxf(rg[3], -MAX_RATIO_F), MAX_RATIO_F);
    float cx = acx + dx * aw, cy = acy + dy * ah;
    float w = aw * expf(dw), h = ah * expf(dh);
    float4 bx;
    bx.x = fminf(fmaxf(cx - 0.5f * w, 0.f), 1.f);
    bx.y = fminf(fmaxf(cy - 0.5f * h, 0.f), 1.f);
    bx.z = fminf(fmaxf(cx + 0.5f * w, 0.f), 1.f);
    bx.w = fminf(fmaxf(cy + 0.5f * h, 0.f), 1.f);
    *(float4*)(boxArr + (size_t)t * 4) = bx;

    // per-class NMS score rows: classes 1..20, contiguous in n for async copy
    float* crow = clsArr + ((size_t)b * NCLS) * NN + n;
    for (int c = 1; c < CC; ++c)
        crow[(size_t)(c - 1) * NN] = valid ? lev[c] : -BIGF;
}

// ---------------------------------------------------------------------------
// Kernel 2: per-(batch,class) soft-NMS scan, scores resident in 80 KB LDS.
// One block per (b,class); 100 sequential steps of argmax + IoU rescoring.
// ---------------------------------------------------------------------------
__global__ __launch_bounds__(NT) void fd_nms(
    const float* __restrict__ clsArr,   // (B,NCLS,N)
    const float* __restrict__ boxArr,   // (B,N,4)
    const float* __restrict__ score,    // (B,N,1)
    const float* __restrict__ logits,   // (B,N,C)
    int*   __restrict__ idxArr,         // (B,NCLS,P)
    float* __restrict__ candArr)        // (B,NCLS,P)
{
    __shared__ __align__(16) float m[NN];          // 80 KB: CDNA5 320KB-LDS path
    __shared__ float redv[NT];
    __shared__ int   redi[NT];
    __shared__ float selb[5];
    __shared__ int   sel_i, sel_valid;

    const int tid = threadIdx.x;
    const int blk = blockIdx.x;          // b*NCLS + ci
    const int b   = blk / NCLS;
    const int ci  = blk - b * NCLS;      // class-1

    const float* src   = clsArr + (size_t)blk * NN;
    const float* boxes = boxArr + (size_t)b * NN * 4;
    __builtin_prefetch(boxes, 0, 1);     // global_prefetch_b8 (L2 warm)

#if defined(__gfx1250__)
    // Async global->LDS fill of the score row (gfx1250 ASYNCcnt path).
    for (int j = tid * 4; j < NN; j += NT * 4) {
        unsigned ldsOff = (unsigned)(size_t)(const void*)&m[j];
        const float* g = src + j;
        asm volatile("global_load_async_to_lds_b128 %0, %1, off"
                     :: "v"(ldsOff), "v"(g) : "memory");
    }
    asm volatile("s_wait_asynccnt 0" ::: "memory");
#else
    for (int j = tid; j < NN; j += NT) m[j] = src[j];
#endif
    __syncthreads();

    for (int p = 0; p < PP; ++p) {
        // ---- argmax over m[], first-index tie break (matches jnp.argmax) ----
        float bv = -INFINITY; int bi = 0;
        for (int j = tid; j < NN; j += NT) {
            float v = m[j];
            if (v > bv) { bv = v; bi = j; }   // strict > keeps lowest j in stride
        }
        redv[tid] = bv; redi[tid] = bi;
        __syncthreads();
        for (int s2 = NT / 2; s2 > 0; s2 >>= 1) {
            if (tid < s2) {
                float ov = redv[tid + s2]; int oi = redi[tid + s2];
                float cv = redv[tid];
                if (ov > cv || (ov == cv && oi < redi[tid])) {
                    redv[tid] = ov; redi[tid] = oi;
                }
            }
            __syncthreads();
        }
        if (tid == 0) {
            int i = redi[0];
            int valid = (redv[0] > 0.0f) ? 1 : 0;
            sel_i = i; sel_valid = valid;
            const float* bxp = boxes + (size_t)i * 4;
            float x1 = bxp[0], y1 = bxp[1], x2 = bxp[2], y2 = bxp[3];
            selb[0] = x1; selb[1] = y1; selb[2] = x2; selb[3] = y2;
            selb[4] = (x2 - x1) * (y2 - y1);
            idxArr[blk * PP + p] = i;
            // cand uses RAW le (score*logit), not the decayed score
            candArr[blk * PP + p] = valid
                ? score[(size_t)b * NN + i] *
                  logits[((size_t)b * NN + i) * CC + (ci + 1)]
                : -BIGF;
        }
        __syncthreads();

        int i = sel_i;
        if (!sel_valid) {
            // max <= 0: every remaining step is invalid in the reference too;
            // its outputs are masked by ok=False, so fill and exit (uniform).
            for (int q = p + 1 + tid; q < PP; q += NT) {
                idxArr[blk * PP + q] = 0;
                candArr[blk * PP + q] = -BIGF;
            }
            break;
        }

        float sx1 = selb[0], sy1 = selb[1], sx2 = selb[2], sy2 = selb[3];
        float sa  = selb[4];
        for (int j = tid; j < NN; j += NT) {
            const float4 bj = *(const float4*)(boxes + (size_t)j * 4);
            float ix1 = fmaxf(sx1, bj.x), iy1 = fmaxf(sy1, bj.y);
            float ix2 = fminf(sx2, bj.z), iy2 = fminf(sy2, bj.w);
            float inter = fmaxf(ix2 - ix1, 0.f) * fmaxf(iy2 - iy1, 0.f);
            float aj  = (bj.z - bj.x) * (bj.w - bj.y);
            float iou = inter / (sa + aj - inter + 1e-9f);
            // w = exp(-0.5*iou^2/SIGMA), SIGMA=0.5  ->  exp(-iou^2)
            float wgt = expf(-iou * iou);
            m[j] = (j == i) ? -BIGF : m[j] * wgt;
        }
        __syncthreads();
    }
}

// ---------------------------------------------------------------------------
// Kernel 3: per-batch top-100 of 2000 candidates (lax.top_k semantics:
// descending, lower index first on ties) + gather of logits/boxes.
// ---------------------------------------------------------------------------
__global__ __launch_bounds__(NT) void fd_topk(
    const float* __restrict__ candArr,  // (B,NCLS,P)
    const int*   __restrict__ idxArr,   // (B,NCLS,P)
    const float* __restrict__ score,    // (B,N,1)
    const float* __restrict__ logits,   // (B,N,C)
    const float* __restrict__ boxArr,   // (B,N,4)
    float* __restrict__ out)            // 8400 logits + 1600 proposals
{
    __shared__ float cv[NCLS * PP];
    __shared__ float redv[NT];
    __shared__ int   redi[NT];
    __shared__ int   selTop[PP];
    __shared__ float selVal[PP];

    const int b = blockIdx.x, tid = threadIdx.x;
    const float* cand = candArr + (size_t)b * NCLS * PP;
    for (int j = tid; j < NCLS * PP; j += NT) cv[j] = cand[j];
    __syncthreads();

    for (int p = 0; p < PP; ++p) {
        float bv = -INFINITY; int bi = 0;
        for (int j = tid; j < NCLS * PP; j += NT) {
            float v = cv[j];
            if (v > bv) { bv = v; bi = j; }
        }
        redv[tid] = bv; redi[tid] = bi;
        __syncthreads();
        for (int s2 = NT / 2; s2 > 0; s2 >>= 1) {
            if (tid < s2) {
                float ov = redv[tid + s2]; int oi = redi[tid + s2];
                if (ov > redv[tid] || (ov == redv[tid] && oi < redi[tid])) {
                    redv[tid] = ov; redi[tid] = oi;
                }
            }
            __syncthreads();
        }
        if (tid == 0) {
            selTop[p] = redi[0];
            selVal[p] = redv[0];
            cv[redi[0]] = -INFINITY;    // distinct from -BIGF: never re-picked
        }
        __syncthreads();
    }

    float* logitOut = out;                         // (B,P,C)
    float* propOut  = out + (size_t)BB * PP * CC;  // (B,P,4)

    for (int t = tid; t < PP * CC; t += NT) {
        int p = t / CC, k = t - p * CC;
        int ti = selTop[p];
        bool ok = selVal[p] > -(BIGF * 0.5f);
        int fi = idxArr[(size_t)b * NCLS * PP + ti];
        float v = ok ? score[(size_t)b * NN + fi] *
                       logits[((size_t)b * NN + fi) * CC + k]
                     : 0.f;
        logitOut[((size_t)b * PP + p) * CC + k] = v;
    }
    for (int t = tid; t < PP * 4; t += NT) {
        int p = t >> 2, k = t & 3;
        int ti = selTop[p];
        bool ok = selVal[p] > -(BIGF * 0.5f);
        int fi = idxArr[(size_t)b * NCLS * PP + ti];
        float v = ok ? boxArr[((size_t)b * NN + fi) * 4 + k] : 0.f;
        propOut[((size_t)b * PP + p) * 4 + k] = v;
    }
}

// ---------------------------------------------------------------------------
extern "C" void kernel_launch(void* const* d_in, const int* in_sizes, int n_in,
                              void* d_out, int out_size, void* d_ws, size_t ws_size,
                              hipStream_t stream) {
    const float* score   = (const float*)d_in[0];   // (B,N,1)
    const float* logits  = (const float*)d_in[1];   // (B,N,C)
    const float* regress = (const float*)d_in[2];   // (B,N,4)
    const float* anchors = (const float*)d_in[3];   // (N,4)
    float* out = (float*)d_out;

    // workspace layout (~7.75 MB)
    float* boxArr  = (float*)d_ws;                         // B*N*4
    float* clsArr  = boxArr + (size_t)BB * NN * 4;         // B*NCLS*N
    int*   idxArr  = (int*)(clsArr + (size_t)BB * NCLS * NN); // B*NCLS*P
    float* candArr = (float*)(idxArr + BB * NCLS * PP);    // B*NCLS*P

    fd_prep<<<(BB * NN + NT - 1) / NT, NT, 0, stream>>>(
        score, logits, regress, anchors, boxArr, clsArr);
    fd_nms<<<BB * NCLS, NT, 0, stream>>>(
        clsArr, boxArr, score, logits, idxArr, candArr);
    fd_topk<<<BB, NT, 0, stream>>>(
        candArr, idxArr, score, logits, boxArr, out);
}